// SinkhornRouter_56435870269502
// MI455X (gfx1250) — compile-verified
//
#include <hip/hip_runtime.h>
#include <stdint.h>

// ---------------------------------------------------------------------------
// Sinkhorn router (S=524288 tokens, E=64 experts, top-2) for MI455X (gfx1250).
//
// q after any number of Sinkhorn steps == exp(logit - rowmax) * r_i * c_j, so
// we keep only r (S floats) and c (64 floats) and stream the logits matrix
// once per iteration.  11 passes x 128 MB ~= 1.5 GB -> ~63 us at 23.3 TB/s;
// logits fit in the 192 MB L2 so passes 2..11 run at L2 bandwidth.
//
// Streaming uses the CDNA5 Tensor Data Mover: each wave DMAs 16x64 fp32
// tiles (4 KB) into a private LDS double buffer via tensor_load_to_lds,
// synchronizes with s_wait_tensorcnt, and computes (exp / row-max / row-dot /
// column partial sums) from LDS while the next tile is in flight.
// ---------------------------------------------------------------------------

#define SINK_ITERS     10
#define SINK_EPS       1e-6f
#define NEXP           64
#define NBLK           512
#define WAVES_PER_BLK  8
#define ROWS_TILE      16
#define TILE_BYTES     (ROWS_TILE * NEXP * 4)          // 4096 B per tile
#define SMEM_BYTES     (WAVES_PER_BLK * 2 * TILE_BYTES) // 64 KB dynamic LDS

typedef unsigned int v4u __attribute__((ext_vector_type(4)));
typedef int          v8i __attribute__((ext_vector_type(8)));
typedef int          v4i __attribute__((ext_vector_type(4)));

__device__ __forceinline__ unsigned rfl_u32(unsigned x) {
  return (unsigned)__builtin_amdgcn_readfirstlane((int)x);
}

// One TDM copy of a ROWS_TILE x 64 fp32 row-major tile (row stride 64 elems)
// from global memory to LDS byte offset lds_off.  Tracked by TENSORcnt.
// D# layout per CDNA5 ISA ch.8 (groups 0/1; groups 2/3 zero => 2D tile).
__device__ __forceinline__ void tdm_load_tile(unsigned lds_off,
                                              const float* gsrc,
                                              unsigned tensor_rows) {
  uint64_t ga   = (uint64_t)(uintptr_t)gsrc;
  unsigned galo = rfl_u32((unsigned)ga);
  unsigned gahi = rfl_u32((unsigned)(ga >> 32)) & 0x01FFFFFFu;  // addr[56:32]
  // group0: {count=1, lds_addr, global_addr_lo, global_addr_hi | type=2}
  v4u g0 = { 1u, rfl_u32(lds_off), galo, gahi | (2u << 30) };
  // group1 bit-exact packing:
  //  w0[17:16]=data_size(2 => 4B); mask/pad/iterate/barrier = 0
  //  w1[31:16]=tensor_dim0.lo16 (64)
  //  w2[15:0 ]=tensor_dim0.hi16 (0), w2[31:16]=tensor_dim1.lo16
  //  w3[15:0 ]=tensor_dim1.hi16,     w3[31:16]=tile_dim0 (64)
  //  w4[15:0 ]=tile_dim1 (ROWS_TILE), w4[31:16]=tile_dim2 (0)
  //  w5       =tensor_dim0_stride.lo32 (64); w6/w7 = 0 (2D)
  v8i g1 = { (int)(2u << 16),
             (int)(64u << 16),
             (int)((tensor_rows & 0xFFFFu) << 16),
             (int)(((tensor_rows >> 16) & 0xFFFFu) | (64u << 16)),
             (int)ROWS_TILE,
             (int)NEXP,
             0, 0 };
  v4i z4 = { 0, 0, 0, 0 };
  __asm__ __volatile__("" ::: "memory");  // keep prior LDS reads above the DMA
#if defined(__clang_major__) && (__clang_major__ >= 23)
  // clang-23 / therock lane: 6-arg form (extra int32x8 descriptor operand)
  v8i z8 = { 0, 0, 0, 0, 0, 0, 0, 0 };
  __builtin_amdgcn_tensor_load_to_lds(g0, g1, z4, z4, z8, 0);
#else
  // ROCm 7.2 / clang-22 lane: 5-arg form
  __builtin_amdgcn_tensor_load_to_lds(g0, g1, z4, z4, 0);
#endif
}

__device__ __forceinline__ void top2_insert(float v, int i, float& v1, int& i1,
                                            float& v2, int& i2) {
  if (v > v1)      { v2 = v1; i2 = i1; v1 = v; i1 = i; }
  else if (v > v2) { v2 = v;  i2 = i; }
}

__device__ __forceinline__ void top2_merge(float& v1, int& i1, float& v2, int& i2,
                                           float w1, int j1, float w2, int j2) {
  if (w1 > v1 || (w1 == v1 && j1 < i1)) {
    float nv2; int ni2;
    if (v1 > w2 || (v1 == w2 && i1 < j2)) { nv2 = v1; ni2 = i1; }
    else                                  { nv2 = w2; ni2 = j2; }
    v1 = w1; i1 = j1; v2 = nv2; i2 = ni2;
  } else if (w1 > v2 || (w1 == v2 && j1 < i2)) {
    v2 = w1; i2 = j1;
  }
}

// ---------------------------------------------------------------------------
// c[j] = 1 initialization
__global__ void sink_init_c(float* __restrict__ c) { c[threadIdx.x] = 1.0f; }

// ---------------------------------------------------------------------------
// Fused pass: row-normalize update of r, plus per-block column partial sums.
__global__ void __launch_bounds__(256)
sinkhorn_pass(const float* __restrict__ logits, float* __restrict__ r,
              const float* __restrict__ c, float* __restrict__ partials,
              int S, int rows_per_wave, int first) {
  extern __shared__ char smem[];
  const int tid  = threadIdx.x;
  const int lane = tid & 31;
  const int wid  = (int)rfl_u32((unsigned)(tid >> 5));
  const int grp  = lane >> 4;   // which row of the pair
  const int cl   = lane & 15;   // column quad (4 floats) index

  const float4 c4 = ((const float4*)c)[cl];

  char*    wbuf     = smem + wid * (2 * TILE_BYTES);
  unsigned wbuf_off = (unsigned)(uintptr_t)wbuf;   // low 32 bits = LDS offset

  const int waveRow0 = (blockIdx.x * WAVES_PER_BLK + wid) * rows_per_wave;
  const int T = rows_per_wave / ROWS_TILE;

  float ax = 0.f, ay = 0.f, az = 0.f, aw = 0.f;    // column accumulators

  tdm_load_tile(wbuf_off, logits + (size_t)waveRow0 * NEXP, (unsigned)S);
  if (T > 1)
    tdm_load_tile(wbuf_off + TILE_BYTES,
                  logits + (size_t)(waveRow0 + ROWS_TILE) * NEXP, (unsigned)S);

  for (int t = 0; t < T; ++t) {
    if (t + 1 < T) __builtin_amdgcn_s_wait_tensorcnt(1);  // oldest tile done
    else           __builtin_amdgcn_s_wait_tensorcnt(0);
    __asm__ __volatile__("" ::: "memory");

    const float4* bp = (const float4*)(wbuf + (t & 1) * TILE_BYTES);
    const int tileRow0 = waveRow0 + t * ROWS_TILE;

#pragma unroll
    for (int j = 0; j < ROWS_TILE / 2; ++j) {
      const int subrow = grp + 2 * j;
      const int row    = tileRow0 + subrow;
      float4 l4 = bp[subrow * 16 + cl];

      float m = fmaxf(fmaxf(l4.x, l4.y), fmaxf(l4.z, l4.w));
      m = fmaxf(m, __shfl_xor(m, 1, 32));
      m = fmaxf(m, __shfl_xor(m, 2, 32));
      m = fmaxf(m, __shfl_xor(m, 4, 32));
      m = fmaxf(m, __shfl_xor(m, 8, 32));

      float qx = expf(l4.x - m), qy = expf(l4.y - m);
      float qz = expf(l4.z - m), qw = expf(l4.w - m);

      float d = qx * c4.x + qy * c4.y + qz * c4.z + qw * c4.w;
      d += __shfl_xor(d, 1, 32);
      d += __shfl_xor(d, 2, 32);
      d += __shfl_xor(d, 4, 32);
      d += __shfl_xor(d, 8, 32);

      float r_old = first ? 1.0f : r[row];
      float r_new = r_old / fmaf(r_old, d, SINK_EPS);
      if (cl == 0) r[row] = r_new;

      ax = fmaf(qx, r_new, ax);
      ay = fmaf(qy, r_new, ay);
      az = fmaf(qz, r_new, az);
      aw = fmaf(qw, r_new, aw);
    }

    if (t + 2 < T)
      tdm_load_tile(wbuf_off + (t & 1) * TILE_BYTES,
                    logits + (size_t)(waveRow0 + (t + 2) * ROWS_TILE) * NEXP,
                    (unsigned)S);
  }

  // Fold the two 16-lane halves (same columns, different rows).
  ax += __shfl_xor(ax, 16, 32);
  ay += __shfl_xor(ay, 16, 32);
  az += __shfl_xor(az, 16, 32);
  aw += __shfl_xor(aw, 16, 32);

  __syncthreads();                       // all streaming done; reuse LDS
  float* red = (float*)smem;             // WAVES_PER_BLK x 64 floats
  if (grp == 0) {
    red[wid * NEXP + cl * 4 + 0] = ax;
    red[wid * NEXP + cl * 4 + 1] = ay;
    red[wid * NEXP + cl * 4 + 2] = az;
    red[wid * NEXP + cl * 4 + 3] = aw;
  }
  __syncthreads();
  if (tid < NEXP) {
    float s = 0.f;
#pragma unroll
    for (int w = 0; w < WAVES_PER_BLK; ++w) s += red[w * NEXP + tid];
    partials[blockIdx.x * NEXP + tid] = s;   // deterministic block partial
  }
}

// ---------------------------------------------------------------------------
// c[j] <- c[j]*col_target / (c[j]*colraw_j + eps); fixed-order reduction.
__global__ void sink_col_update(const float* __restrict__ partials,
                                float* __restrict__ c, int nblk,
                                float col_target) {
  const int j = threadIdx.x;  // 64 threads
  float s = 0.f;
  for (int b = 0; b < nblk; ++b) s += partials[b * NEXP + j];
  const float cj = c[j];
  c[j] = cj * col_target / fmaf(cj, s, SINK_EPS);
}

// ---------------------------------------------------------------------------
// Final row-normalize + top-2 (stable: lowest index wins ties) + renormalize.
__global__ void __launch_bounds__(256)
sinkhorn_topk(const float* __restrict__ logits, const float* __restrict__ r,
              const float* __restrict__ c, long long* __restrict__ out_idx,
              float* __restrict__ out_w, int S, int rows_per_wave) {
  extern __shared__ char smem[];
  const int tid  = threadIdx.x;
  const int lane = tid & 31;
  const int wid  = (int)rfl_u32((unsigned)(tid >> 5));
  const int grp  = lane >> 4;
  const int cl   = lane & 15;

  const float4 c4 = ((const float4*)c)[cl];

  char*    wbuf     = smem + wid * (2 * TILE_BYTES);
  unsigned wbuf_off = (unsigned)(uintptr_t)wbuf;

  const int waveRow0 = (blockIdx.x * WAVES_PER_BLK + wid) * rows_per_wave;
  const int T = rows_per_wave / ROWS_TILE;

  tdm_load_tile(wbuf_off, logits + (size_t)waveRow0 * NEXP, (unsigned)S);
  if (T > 1)
    tdm_load_tile(wbuf_off + TILE_BYTES,
                  logits + (size_t)(waveRow0 + ROWS_TILE) * NEXP, (unsigned)S);

  for (int t = 0; t < T; ++t) {
    if (t + 1 < T) __builtin_amdgcn_s_wait_tensorcnt(1);
    else           __builtin_amdgcn_s_wait_tensorcnt(0);
    __asm__ __volatile__("" ::: "memory");

    const float4* bp = (const float4*)(wbuf + (t & 1) * TILE_BYTES);
    const int tileRow0 = waveRow0 + t * ROWS_TILE;

#pragma unroll
    for (int j = 0; j < ROWS_TILE / 2; ++j) {
      const int subrow = grp + 2 * j;
      const int row    = tileRow0 + subrow;
      float4 l4 = bp[subrow * 16 + cl];

      float m = fmaxf(fmaxf(l4.x, l4.y), fmaxf(l4.z, l4.w));
      m = fmaxf(m, __shfl_xor(m, 1, 32));
      m = fmaxf(m, __shfl_xor(m, 2, 32));
      m = fmaxf(m, __shfl_xor(m, 4, 32));
      m = fmaxf(m, __shfl_xor(m, 8, 32));

      float vx = expf(l4.x - m) * c4.x, vy = expf(l4.y - m) * c4.y;
      float vz = expf(l4.z - m) * c4.z, vw = expf(l4.w - m) * c4.w;

      float d = vx + vy + vz + vw;
      d += __shfl_xor(d, 1, 32);
      d += __shfl_xor(d, 2, 32);
      d += __shfl_xor(d, 4, 32);
      d += __shfl_xor(d, 8, 32);

      const float ri  = r[row];
      const float inv = ri / fmaf(ri, d, SINK_EPS);   // p = v * inv
      float p0 = vx * inv, p1 = vy * inv, p2 = vz * inv, p3 = vw * inv;

      float v1 = p0; int i1 = cl * 4;
      float v2 = -__builtin_inff(); int i2 = 0x7FFFFFFF;
      top2_insert(p1, cl * 4 + 1, v1, i1, v2, i2);
      top2_insert(p2, cl * 4 + 2, v1, i1, v2, i2);
      top2_insert(p3, cl * 4 + 3, v1, i1, v2, i2);

#pragma unroll
      for (int msk = 1; msk <= 8; msk <<= 1) {
        float w1 = __shfl_xor(v1, msk, 32);
        int   j1 = __shfl_xor(i1, msk, 32);
        float w2 = __shfl_xor(v2, msk, 32);
        int   j2 = __shfl_xor(i2, msk, 32);
        top2_merge(v1, i1, v2, i2, w1, j1, w2, j2);
      }

      if (cl == 0) {
        const float invw = 1.0f / (v1 + v2 + SINK_EPS);
        const size_t o = (size_t)row * 2;
        out_idx[o]     = (long long)i1;
        out_idx[o + 1] = (long long)i2;
        out_w[o]       = v1 * invw;
        out_w[o + 1]   = v2 * invw;
      }
    }

    if (t + 2 < T)
      tdm_load_tile(wbuf_off + (t & 1) * TILE_BYTES,
                    logits + (size_t)(waveRow0 + (t + 2) * ROWS_TILE) * NEXP,
                    (unsigned)S);
  }
}

// ---------------------------------------------------------------------------
extern "C" void kernel_launch(void* const* d_in, const int* in_sizes, int n_in,
                              void* d_out, int out_size, void* d_ws,
                              size_t ws_size, hipStream_t stream) {
  (void)n_in; (void)out_size; (void)ws_size;
  const float* logits = (const float*)d_in[0];
  const int S = in_sizes[0] / NEXP;           // 524288

  // workspace: c[64] | partials[NBLK*64] | r[S]   (~2.2 MB)
  float* c        = (float*)d_ws;
  float* partials = c + NEXP;
  float* r        = partials + (size_t)NBLK * NEXP;

  long long* out_idx = (long long*)d_out;
  float*     out_w   = (float*)((char*)d_out +
                                (size_t)S * 2 * sizeof(long long));

  const int   rows_per_wave = S / (NBLK * WAVES_PER_BLK);  // 128
  const float col_target    = (float)S / (float)NEXP;      // 8192

  sink_init_c<<<1, NEXP, 0, stream>>>(c);
  for (int it = 0; it < SINK_ITERS; ++it) {
    sinkhorn_pass<<<NBLK, 256, SMEM_BYTES, stream>>>(
        logits, r, c, partials, S, rows_per_wave, it == 0 ? 1 : 0);
    sink_col_update<<<1, NEXP, 0, stream>>>(partials, c, NBLK, col_target);
  }
  sinkhorn_topk<<<NBLK, 256, SMEM_BYTES, stream>>>(
      logits, r, c, out_idx, out_w, S, rows_per_wave);
}